// transfusion_50611894616901
// MI455X (gfx1250) — compile-verified
//
#include <hip/hip_runtime.h>

// ---------------- problem constants (from reference) ----------------
#define NPTS   50000
#define FCH    2714          // f = int(2 * N^(2/3))
#define MCH    18            // N // f
#define NCHUNK 19            // 18 full chunks + remainder (1148)
#define NPAD   50016         // N + 16, a multiple of 16 for WMMA tiles (one extra zero tile)
#define EPSF   1e-5f

typedef float v2f __attribute__((ext_vector_type(2)));
typedef float v8f __attribute__((ext_vector_type(8)));

__device__ __forceinline__ int chunk_size(int cid) {
  return (cid < MCH) ? FCH : (NPTS - MCH * FCH);   // 2714 or 1148
}

// ---------------- 1. Wf = Wv[128x512] @ Wo[512x128] ----------------
__global__ __launch_bounds__(128)
void k_wfuse(const float* __restrict__ Wv, const float* __restrict__ Wo,
             float* __restrict__ Wf) {
  const int r = blockIdx.x, c = threadIdx.x;
  float acc = 0.f;
#pragma unroll 8
  for (int k = 0; k < 512; ++k)
    acc = fmaf(Wv[r * 512 + k], Wo[k * 128 + c], acc);
  Wf[r * 128 + c] = acc;
}

// ---------------- 2. per-chunk top-5 of sim columns ----------------
// sim[i,j] = x_c[i] . h_c[j] ; top over i for each j (chunk-local).
// Strict '>' displacement == stable tie-break (lower index wins), matching top_k.
__global__ __launch_bounds__(256)
void k_topk(const float* __restrict__ xc, const float* __restrict__ hc,
            int* __restrict__ tidx) {
  __shared__ float xs[256 * 3];
  const int cid  = blockIdx.y;
  const int base = cid * FCH;
  const int fc   = chunk_size(cid);
  const int j    = blockIdx.x * 256 + threadIdx.x;
  const bool valid = (j < fc);
  float hx = 0.f, hy = 0.f, hz = 0.f;
  if (valid) {
    const float* hp = hc + (size_t)(base + j) * 3;
    hx = hp[0]; hy = hp[1]; hz = hp[2];
  }
  float v0 = -INFINITY, v1 = -INFINITY, v2 = -INFINITY, v3 = -INFINITY, v4 = -INFINITY;
  int   i0 = 0, i1 = 0, i2 = 0, i3 = 0, i4 = 0;

  for (int t0 = 0; t0 < fc; t0 += 256) {
    const int nt = min(256, fc - t0);
    __syncthreads();
    for (int q = threadIdx.x; q < nt * 3; q += 256)
      xs[q] = xc[(size_t)(base + t0) * 3 + q];
    __syncthreads();
    if (valid) {
      for (int ii = 0; ii < nt; ++ii) {
        const float d = fmaf(xs[3 * ii], hx, fmaf(xs[3 * ii + 1], hy, xs[3 * ii + 2] * hz));
        if (d > v4) {
          const int idx = t0 + ii;
          if (d > v2) {
            if (d > v1) {
              if (d > v0) { v4=v3;i4=i3; v3=v2;i3=i2; v2=v1;i2=i1; v1=v0;i1=i0; v0=d;i0=idx; }
              else        { v4=v3;i4=i3; v3=v2;i3=i2; v2=v1;i2=i1; v1=d;i1=idx; }
            } else        { v4=v3;i4=i3; v3=v2;i3=i2; v2=d;i2=idx; }
          } else {
            if (d > v3)   { v4=v3;i4=i3; v3=d;i3=idx; }
            else          { v4=d;i4=idx; }
          }
        }
      }
    }
  }
  if (valid) {
    int* tp = tidx + (size_t)(base + j) * 5;
    tp[0] = i0; tp[1] = i1; tp[2] = i2; tp[3] = i3; tp[4] = i4;
  }
}

// ---------------- 3. scrambled gather + row sum ----------------
// s[i,:] = x_f[i,:] + sum_t h_f[ base + tidx[(p%fc), (p/fc)] , :],  p = il*5+t
__global__ __launch_bounds__(256)
void k_gather(const float* __restrict__ xf, const float* __restrict__ hf,
              const int* __restrict__ tidx, float* __restrict__ sbuf) {
  const int lane = threadIdx.x & 31;
  const int wid  = threadIdx.x >> 5;
  const long row = (long)blockIdx.x * 8 + wid;
  if (row >= NPAD) return;
  float* sp = sbuf + row * 128;
  if (row >= NPTS) {           // zero padding rows so WMMA tiles read zeros
#pragma unroll
    for (int u = 0; u < 4; ++u) sp[lane + 32 * u] = 0.f;
    return;
  }
  const int cid  = (int)(row / FCH);          // rows >= 18*f land in remainder chunk 18
  const int base = cid * FCH;
  const int fc   = chunk_size(cid);
  const int il   = (int)row - base;
  const float* xp = xf + row * 128;
  float a0 = xp[lane], a1 = xp[lane + 32], a2 = xp[lane + 64], a3 = xp[lane + 96];
#pragma unroll
  for (int t = 0; t < 5; ++t) {
    const int p  = il * 5 + t;
    const int jj = p % fc;
    const int kk = p / fc;
    const long nb = (long)base + tidx[(size_t)(base + jj) * 5 + kk];
    const float* hp = hf + nb * 128;
    a0 += hp[lane]; a1 += hp[lane + 32]; a2 += hp[lane + 64]; a3 += hp[lane + 96];
  }
  sp[lane] = a0; sp[lane + 32] = a1; sp[lane + 64] = a2; sp[lane + 96] = a3;
}

// ---------------- 4/7. fp32 WMMA GEMM: out[NPADx128] = A[NPADx128]*W[128x128]+bias ----------------
// One wave = one 16x16 tile, K=128 via 32x V_WMMA_F32_16X16X4_F32.
// A frag (ISA layout): lanes 0-15 -> row m0+l16, K {4kk,4kk+1}; lanes 16-31 -> K {4kk+2,4kk+3}.
// B frag mirrors A: lanes 0-15 -> col n0+l16, K {4kk,4kk+1}; lanes 16-31 -> K {4kk+2,4kk+3}.
// C/D: VGPR v -> row m0+v (lanes 0-15) / m0+v+8 (lanes 16-31), col n0+l16.
__global__ __launch_bounds__(256)
void k_gemm128(const float* __restrict__ A, const float* __restrict__ W,
               const float* __restrict__ bias, float* __restrict__ out) {
  const int lane = threadIdx.x & 31;
  const int wid  = threadIdx.x >> 5;          // 0..7 -> n-tile
  const int m0   = blockIdx.x * 16;
  const int n0   = wid * 16;
  const int hlf  = lane >> 4;                 // lane half
  const int l16  = lane & 15;

  v8f acc = {};
  const float* arow = A + (size_t)(m0 + l16) * 128 + 2 * hlf;
  const float* bcol = W + (size_t)(2 * hlf) * 128 + (n0 + l16);
#pragma unroll
  for (int kk = 0; kk < 32; ++kk) {
    v2f a = *(const v2f*)(arow + 4 * kk);              // 8B aligned (even index)
    v2f b;
    b.x = bcol[4 * kk * 128];
    b.y = bcol[4 * kk * 128 + 128];
    acc = __builtin_amdgcn_wmma_f32_16x16x4_f32(false, a, false, b,
                                                (short)0, acc, false, false);
  }
  const float bv = bias[n0 + l16];
#pragma unroll
  for (int v = 0; v < 8; ++v)
    out[(size_t)(m0 + v + 8 * hlf) * 128 + n0 + l16] = acc[v] + bv;
}

// ---------------- 5. per-chunk column mean / rstd ----------------
__global__ __launch_bounds__(512)
void k_chunk_stats(const float* __restrict__ out, float* __restrict__ mu,
                   float* __restrict__ rstd) {
  __shared__ float ls[512], lq[512];
  const int cid  = blockIdx.x;
  const int base = cid * FCH;
  const int fc   = chunk_size(cid);
  const int col  = threadIdx.x & 127;
  const int seg  = threadIdx.x >> 7;          // 0..3
  float s = 0.f, q = 0.f;
  for (int r = seg; r < fc; r += 4) {
    const float v = out[(size_t)(base + r) * 128 + col];
    s += v; q = fmaf(v, v, q);
  }
  ls[threadIdx.x] = s; lq[threadIdx.x] = q;
  __syncthreads();
  if (seg == 0) {
    const float S = ls[col] + ls[col + 128] + ls[col + 256] + ls[col + 384];
    const float Q = lq[col] + lq[col + 128] + lq[col + 256] + lq[col + 384];
    const float m = S / (float)fc;
    float var = Q / (float)fc - m * m;
    if (var < 0.f) var = 0.f;
    mu[cid * 128 + col]   = m;
    rstd[cid * 128 + col] = rsqrtf(var + EPSF);
  }
}

// ---------------- 6. bn_a + relu -> sbuf (input of 2nd GEMM) ----------------
__global__ __launch_bounds__(256)
void k_apply_a(const float* __restrict__ out, const float* __restrict__ mu,
               const float* __restrict__ rstd, const float* __restrict__ gamma,
               const float* __restrict__ beta, float* __restrict__ sbuf) {
  const long idx = (long)blockIdx.x * 256 + threadIdx.x;
  const long row = idx >> 7;
  const int  col = (int)(idx & 127);
  if (row >= NPAD) return;
  if (row >= NPTS) { sbuf[idx] = 0.f; return; }
  const int cid = (int)(row / FCH);
  const float v = out[idx];
  const float r = fmaf(gamma[col] * (v - mu[cid * 128 + col]),
                       rstd[cid * 128 + col], beta[col]);
  sbuf[idx] = r > 0.f ? r : 0.f;
}

// ---------------- 8a. global stats, stage 1 (partial sums, fixed order) ----------------
__global__ __launch_bounds__(128)
void k_gstats1(const float* __restrict__ y, float* __restrict__ partS,
               float* __restrict__ partQ) {
  const int col = threadIdx.x;
  float s = 0.f, q = 0.f;
  for (int r = blockIdx.x; r < NPTS; r += 256) {
    const float v = y[(size_t)r * 128 + col];
    s += v; q = fmaf(v, v, q);
  }
  partS[blockIdx.x * 128 + col] = s;
  partQ[blockIdx.x * 128 + col] = q;
}

// ---------------- 8b. global stats, stage 2 ----------------
__global__ __launch_bounds__(128)
void k_gstats2(const float* __restrict__ partS, const float* __restrict__ partQ,
               float* __restrict__ gmu, float* __restrict__ grstd) {
  const int col = threadIdx.x;
  float S = 0.f, Q = 0.f;
  for (int b = 0; b < 256; ++b) { S += partS[b * 128 + col]; Q += partQ[b * 128 + col]; }
  const float m = S / (float)NPTS;
  float var = Q / (float)NPTS - m * m;
  if (var < 0.f) var = 0.f;
  gmu[col]   = m;
  grstd[col] = rsqrtf(var + EPSF);
}

// ---------------- 9. final bn_p + relu -> d_out ----------------
__global__ __launch_bounds__(256)
void k_final(const float* __restrict__ y, const float* __restrict__ gmu,
             const float* __restrict__ grstd, const float* __restrict__ gamma,
             const float* __restrict__ beta, float* __restrict__ dst) {
  const long idx = (long)blockIdx.x * 256 + threadIdx.x;  // < NPTS*128 exactly
  const int col = (int)(idx & 127);
  const float v = y[idx];
  const float r = fmaf(gamma[col] * (v - gmu[col]), grstd[col], beta[col]);
  dst[idx] = r > 0.f ? r : 0.f;
}

// ---------------- host launch ----------------
extern "C" void kernel_launch(void* const* d_in, const int* in_sizes, int n_in,
                              void* d_out, int out_size, void* d_ws, size_t ws_size,
                              hipStream_t stream) {
  (void)in_sizes; (void)n_in; (void)out_size; (void)ws_size;
  const float* h_f  = (const float*)d_in[0];
  const float* x_f  = (const float*)d_in[1];
  const float* x_c  = (const float*)d_in[2];
  const float* h_c  = (const float*)d_in[3];
  // d_in[4] = Wq, d_in[5] = Wk : dead (softmax over size-1 axis -> all-ones weights)
  const float* Wv   = (const float*)d_in[6];
  const float* Wo   = (const float*)d_in[7];
  const float* bo   = (const float*)d_in[8];
  const float* g_a  = (const float*)d_in[9];
  const float* b_a  = (const float*)d_in[10];
  const float* Wp   = (const float*)d_in[11];
  const float* bp   = (const float*)d_in[12];
  const float* g_p  = (const float*)d_in[13];
  const float* b_p  = (const float*)d_in[14];

  // workspace layout (256B aligned, ~52.6 MB total)
  char* w = (char*)d_ws;
  size_t off = 0;
  auto take = [&](size_t bytes) { char* p = w + off; off = (off + bytes + 255) & ~(size_t)255; return p; };
  float* Wf    = (float*)take(128 * 128 * 4);
  int*   tidx  = (int*)  take((size_t)NPTS * 5 * 4);
  float* sbuf  = (float*)take((size_t)NPAD * 128 * 4);
  float* obuf  = (float*)take((size_t)NPAD * 128 * 4);
  float* mu_a  = (float*)take(NCHUNK * 128 * 4);
  float* rs_a  = (float*)take(NCHUNK * 128 * 4);
  float* partS = (float*)take(256 * 128 * 4);
  float* partQ = (float*)take(256 * 128 * 4);
  float* gmu   = (float*)take(128 * 4);
  float* grs   = (float*)take(128 * 4);

  k_wfuse<<<dim3(128), dim3(128), 0, stream>>>(Wv, Wo, Wf);
  k_topk<<<dim3((FCH + 255) / 256, NCHUNK), dim3(256), 0, stream>>>(x_c, h_c, tidx);
  k_gather<<<dim3(NPAD / 8), dim3(256), 0, stream>>>(x_f, h_f, tidx, sbuf);
  k_gemm128<<<dim3(NPAD / 16), dim3(256), 0, stream>>>(sbuf, Wf, bo, obuf);
  k_chunk_stats<<<dim3(NCHUNK), dim3(512), 0, stream>>>(obuf, mu_a, rs_a);
  k_apply_a<<<dim3((NPAD * 128) / 256), dim3(256), 0, stream>>>(obuf, mu_a, rs_a, g_a, b_a, sbuf);
  k_gemm128<<<dim3(NPAD / 16), dim3(256), 0, stream>>>(sbuf, Wp, bp, obuf);
  k_gstats1<<<dim3(256), dim3(128), 0, stream>>>(obuf, partS, partQ);
  k_gstats2<<<dim3(1), dim3(128), 0, stream>>>(partS, partQ, gmu, grs);
  k_final<<<dim3((NPTS * 128) / 256), dim3(256), 0, stream>>>(obuf, gmu, grs, g_p, b_p, (float*)d_out);
}